// ProposalLayer_50182397887268
// MI455X (gfx1250) — compile-verified
//
#include <hip/hip_runtime.h>
#include <hip/hip_bf16.h>

// ProposalLayer for MI455X (gfx1250, wave32).
// dist^2(p,g) = |r_g|^2 + |c_p|^2 - 2 r_g . c_p  computed via
// V_WMMA_F32_16X16X4_F32 with augmented K=4:
//   A row g = [-2rx, -2ry, -2rz, |r|^2]   (invalid g -> [0,0,0,1e30])
//   B col p = [ cx,   cy,   cz,  1    ]
// D[g,p] = |r|^2 - 2 r.c ; argmin over g unaffected by the per-column |c|^2.

typedef __attribute__((ext_vector_type(2))) float v2f;
typedef __attribute__((ext_vector_type(8))) float v8f;

#define NB 4096
#define NP 128
#define NG 128

__global__ __launch_bounds__(256) void proposal_layer_kernel(
    const int*   __restrict__ topk_index,   // [B,P,3]
    const float* __restrict__ topk_confs,   // [B,P]
    const float* __restrict__ match_bbox,   // [B,P,2]
    const float* __restrict__ roots_3d,     // [B,G,3]
    const float* __restrict__ gt_bbox,      // [B,G,2]
    const int*   __restrict__ num_person,   // [B]
    float*       __restrict__ out)          // [B,P,7]
{
    const int b      = blockIdx.x;
    const int wave   = threadIdx.x >> 5;     // 0..7  -> p-tile of 16
    const int lane   = threadIdx.x & 31;
    const int lane16 = lane & 15;
    const bool hi    = lane >= 16;

    // voxel -> world mm
    const float SX = 8000.0f / 79.0f, SY = 8000.0f / 79.0f, SZ = 2000.0f / 19.0f;
    const float BX = -4000.0f,        BY = -4000.0f,        BZ = 0.0f;

    const int p = wave * 16 + lane16;        // both lane halves mirror the same p
    const int* ti = topk_index + ((size_t)b * NP + p) * 3;
    const float cx = (float)ti[0] * SX + BX;
    const float cy = (float)ti[1] * SY + BY;
    const float cz = (float)ti[2] * SZ + BZ;
    const float cnorm2 = cx * cx + cy * cy + cz * cz;

    // B operand (4x16 f32): lanes 0-15 hold K=0,1 ; lanes 16-31 hold K=2,3
    v2f bm;
    bm.x = hi ? cz   : cx;
    bm.y = hi ? 1.0f : cy;

    const int np = num_person[b];

    float minval = 3.0e38f;  // running min of (|r|^2 - 2 r.c)
    int   ming   = 0;

    #pragma unroll
    for (int t = 0; t < 8; ++t) {
        const int g = t * 16 + lane16;       // A row handled by this lane
        const float* r = roots_3d + ((size_t)b * NG + g) * 3;
        const float rx = r[0], ry = r[1], rz = r[2];
        const bool valid = g < np;
        // invalid gt row -> [0,0,0,1e30] (avoid inf: 0*inf -> NaN in WMMA)
        const float rn  = valid ? (rx * rx + ry * ry + rz * rz) : 1.0e30f;
        const float m2x = valid ? -2.0f * rx : 0.0f;
        const float m2y = valid ? -2.0f * ry : 0.0f;
        const float m2z = valid ? -2.0f * rz : 0.0f;

        v2f am;
        am.x = hi ? m2z : m2x;
        am.y = hi ? rn  : m2y;

        v8f cacc = {};
        // (neg_a, A, neg_b, B, c_mod, C, reuse_a, reuse_b)
        v8f d = __builtin_amdgcn_wmma_f32_16x16x4_f32(
            false, am, false, bm, (short)0, cacc, false, false);

        // D layout: lane = column p (lane&15); VGPR v = row g = t*16 + v (+8 on hi half)
        const int gbase = t * 16 + (hi ? 8 : 0);
        #pragma unroll
        for (int v = 0; v < 8; ++v) {
            const float val = d[v];
            const int   gi  = gbase + v;
            const bool  lt  = val < minval;   // strict: keeps first (lowest g)
            minval = lt ? val : minval;
            ming   = lt ? gi  : ming;
        }
    }

    // combine lane-halves (half0 owns g%16 in 0..7, half1 owns 8..15)
    const float oval = __shfl_xor(minval, 16, 32);
    const int   og   = __shfl_xor(ming,   16, 32);
    if (oval < minval || (oval == minval && og < ming)) {
        minval = oval;
        ming   = og;
    }

    // epilogue: lanes 0-15 each own one proposal p
    if (!hi) {
        const float d2 = minval + cnorm2;                 // min dist^2
        const float p2gt = (d2 > 500.0f * 500.0f) ? -1.0f : (float)ming;

        const float conf = topk_confs[(size_t)b * NP + p];
        const float* mb  = match_bbox + ((size_t)b * NP + p) * 2;
        const float pb0 = mb[0], pb1 = mb[1];

        const int safe = (p2gt < 0.0f) ? 0 : ming;        // clip(-1..) -> 0
        const float* gb = gt_bbox + ((size_t)b * NG + safe) * 2;
        const float g0 = gb[0], g1 = gb[1];

        const bool ow = (p2gt >= 0.0f) &&
                        ((pb0 < g0 - 0.1f) || (pb1 < g1 - 0.1f));
        const float o5 = ow ? g0 : pb0;
        const float o6 = ow ? g1 : pb1;

        float* o = out + ((size_t)b * NP + p) * 7;
        o[0] = cx; o[1] = cy; o[2] = cz;
        o[3] = p2gt; o[4] = conf; o[5] = o5; o[6] = o6;
    }
}

extern "C" void kernel_launch(void* const* d_in, const int* in_sizes, int n_in,
                              void* d_out, int out_size, void* d_ws, size_t ws_size,
                              hipStream_t stream) {
    (void)in_sizes; (void)n_in; (void)d_ws; (void)ws_size; (void)out_size;
    const int*   topk_index = (const int*)  d_in[0];
    const float* topk_confs = (const float*)d_in[1];
    const float* match_bbox = (const float*)d_in[2];
    const float* roots_3d   = (const float*)d_in[3];
    const float* gt_bbox    = (const float*)d_in[4];
    const int*   num_person = (const int*)  d_in[5];
    float* out = (float*)d_out;

    dim3 grid(NB), block(256);  // 8 waves/block: one p-tile each; one block/batch
    proposal_layer_kernel<<<grid, block, 0, stream>>>(
        topk_index, topk_confs, match_bbox, roots_3d, gt_bbox, num_person, out);
}